// AdaptiveECE_635655159836
// MI455X (gfx1250) — compile-verified
//
#include <hip/hip_runtime.h>
#include <cmath>

// ---------------------------------------------------------------------------
// Adaptive ECE for MI455X (gfx1250, wave32)
// Phase 1: wave-per-row two-pass softmax-max reduction (memory-bound, ~1 GB read)
// Phase 2: hi-16-bit histogram of conf bit patterns (monotone key, conf>0)
// Phase 3: locate buckets for the ~30 required order-statistic ranks
// Phase 4: per-rank 8+8-bit refinement -> exact order statistic values
// Phase 5: assemble 16 interp'd quantile edges (np.interp f32 semantics)
// Phase 6: bin samples, per-block partial sums (int counts/acc, f64 conf)
// Phase 7: exact f32 WMMA (V_WMMA_F32_16X16X4_F32) reduction of partials + ECE
// ---------------------------------------------------------------------------

#define NBINS 15
#define NTGT  30
#define NB_PART 512   // number of partial-producing blocks in phase 6

typedef __attribute__((ext_vector_type(2))) float v2f;
typedef __attribute__((ext_vector_type(8))) float v8f;

struct RankParams { int r[NTGT]; };
struct FracParams { float f[16]; };

// ---------------------------------------------------------------- Phase 1 ---
// One wave (32 lanes) per row. Two-pass over registers: max/argmax, then
// sum(exp(x - max)). conf = 1/sum. Coalesced float4 loads (global_load_b128).
__global__ void __launch_bounds__(256)
row_reduce_kernel(const float* __restrict__ logits, const int* __restrict__ labels,
                  float* __restrict__ conf, unsigned* __restrict__ acc,
                  int N, int C) {
    const int wave = threadIdx.x >> 5;
    const int lane = threadIdx.x & 31;
    const int row  = blockIdx.x * 8 + wave;
    if (row >= N) return;

    const float* rp = logits + (size_t)row * (size_t)C;

    float4 v[8];
#pragma unroll
    for (int j = 0; j < 8; ++j) {
        int idx = j * 128 + lane * 4;
        if (idx + 4 <= C) {
            v[j] = *(const float4*)(rp + idx);
        } else {
            v[j] = make_float4(-INFINITY, -INFINITY, -INFINITY, -INFINITY);
        }
    }

    // pass A: per-lane max + first-occurrence argmax
    float m = -INFINITY;
    int   mi = 0x7fffffff;
#pragma unroll
    for (int j = 0; j < 8; ++j) {
        float xs[4] = {v[j].x, v[j].y, v[j].z, v[j].w};
#pragma unroll
        for (int e = 0; e < 4; ++e) {
            int idx = j * 128 + lane * 4 + e;
            if (xs[e] > m) { m = xs[e]; mi = idx; }
        }
    }
    // wave32 butterfly reduce (lowest index wins ties, matching jnp.argmax)
#pragma unroll
    for (int off = 16; off > 0; off >>= 1) {
        float om = __shfl_xor(m, off, 32);
        int   oi = __shfl_xor(mi, off, 32);
        if (om > m || (om == m && oi < mi)) { m = om; mi = oi; }
    }

    // pass B: sum of exp(x - max); padded -INF lanes contribute exp(-inf)=0
    float s = 0.0f;
#pragma unroll
    for (int j = 0; j < 8; ++j) {
        float xs[4] = {v[j].x, v[j].y, v[j].z, v[j].w};
#pragma unroll
        for (int e = 0; e < 4; ++e) s += expf(xs[e] - m);
    }
#pragma unroll
    for (int off = 16; off > 0; off >>= 1) s += __shfl_xor(s, off, 32);

    if (lane == 0) {
        conf[row] = 1.0f / s;                       // max softmax prob
        acc[row]  = (mi == labels[row]) ? 1u : 0u;  // prediction correct?
    }
}

// ---------------------------------------------------------------- Phase 2 ---
__global__ void hist_hi_kernel(const float* __restrict__ conf,
                               unsigned* __restrict__ hist, int N) {
    int i = blockIdx.x * blockDim.x + threadIdx.x;
    if (i < N) {
        unsigned key = __float_as_uint(conf[i]); // conf in (0,1] -> monotone bits
        atomicAdd(&hist[key >> 16], 1u);
    }
}

// ---------------------------------------------------------------- Phase 3 ---
// One block of 1024 threads scans the 65536-bin histogram (64 bins/thread),
// block-scans totals, and locates the hi-16 bucket + local rank for each of
// the NTGT required order-statistic ranks.
__global__ void __launch_bounds__(1024)
find_buckets_kernel(const unsigned* __restrict__ hist, RankParams rp,
                    unsigned* __restrict__ tBucket, unsigned* __restrict__ tLrank) {
    __shared__ unsigned sc[1024];
    const int t = threadIdx.x;

    unsigned s = 0;
#pragma unroll 4
    for (int j = 0; j < 64; ++j) s += hist[t * 64 + j];

    sc[t] = s;
    __syncthreads();
    for (int off = 1; off < 1024; off <<= 1) {        // Hillis-Steele inclusive
        unsigned vv = (t >= off) ? sc[t - off] : 0u;
        __syncthreads();
        sc[t] += vv;
        __syncthreads();
    }
    const unsigned base = sc[t] - s;                  // exclusive prefix

    for (int idx = 0; idx < NTGT; ++idx) {
        unsigned r = (unsigned)rp.r[idx];
        if (r >= base && r < base + s) {
            unsigned run = base;
            for (int j = 0; j < 64; ++j) {
                unsigned c = hist[t * 64 + j];
                if (r < run + c) {
                    tBucket[idx] = (unsigned)(t * 64 + j);
                    tLrank[idx]  = r - run;
                    break;
                }
                run += c;
            }
        }
    }
}

// ---------------------------------------------------------------- Phase 4 ---
// One block per target rank: two 8-bit refinement passes over all N keys
// recover the exact 32-bit key of the order statistic.
__global__ void __launch_bounds__(256)
refine_kernel(const float* __restrict__ conf, int N,
              const unsigned* __restrict__ tBucket, const unsigned* __restrict__ tLrank,
              float* __restrict__ tVal) {
    __shared__ unsigned h[256];
    __shared__ unsigned sb, slr, sb2, slr2;
    const int t = threadIdx.x;
    if (t == 0) { sb = tBucket[blockIdx.x]; slr = tLrank[blockIdx.x]; }
    h[t] = 0;
    __syncthreads();
    const unsigned b = sb, lr = slr;

    for (int i = t; i < N; i += 256) {
        unsigned key = __float_as_uint(conf[i]);
        if ((key >> 16) == b) atomicAdd(&h[(key >> 8) & 0xFFu], 1u);
    }
    __syncthreads();
    if (t == 0) {
        unsigned run = 0;
        for (int j = 0; j < 256; ++j) {
            unsigned c = h[j];
            if (lr < run + c) { sb2 = (unsigned)j; slr2 = lr - run; break; }
            run += c;
        }
    }
    __syncthreads();
    const unsigned b2 = sb2, lr2 = slr2;
    h[t] = 0;
    __syncthreads();
    const unsigned hi24 = (b << 8) | b2;

    for (int i = t; i < N; i += 256) {
        unsigned key = __float_as_uint(conf[i]);
        if ((key >> 8) == hi24) atomicAdd(&h[key & 0xFFu], 1u);
    }
    __syncthreads();
    if (t == 0) {
        unsigned run = 0;
        for (int j = 0; j < 256; ++j) {
            unsigned c = h[j];
            if (lr2 < run + c) {
                tVal[blockIdx.x] = __uint_as_float((b << 16) | (b2 << 8) | (unsigned)j);
                break;
            }
            run += c;
        }
    }
}

// ---------------------------------------------------------------- Phase 5 ---
__global__ void make_edges_kernel(const float* __restrict__ tVal,
                                  float* __restrict__ edges, FracParams fp) {
    if (threadIdx.x == 0) {
        edges[0] = tVal[0];                       // sorted[0]
        for (int i = 1; i < NBINS; ++i) {
            float v0 = tVal[2 * i - 1], v1 = tVal[2 * i];
            edges[i] = v0 + (v1 - v0) * fp.f[i];  // np.interp, f32
        }
        edges[NBINS] = tVal[NTGT - 1];            // sorted[N-1]
    }
}

// ---------------------------------------------------------------- Phase 6 ---
// Bin each sample (torch semantics: bin i holds (edges[i], edges[i+1]], conf
// <= edges[0] dropped) and produce deterministic per-block partials:
// [counts 0..15 | acc 16..31 | conf 32..47] as f32 (counts/acc exact ints).
__global__ void __launch_bounds__(256)
bin_accum_kernel(const float* __restrict__ conf, const unsigned* __restrict__ acc,
                 const float* __restrict__ edges, int N,
                 float* __restrict__ partials) {
    __shared__ float   e[16];
    __shared__ unsigned cnt[16], accs[16];
    __shared__ double  cfs[16];
    const int t = threadIdx.x;
    if (t < 16) { e[t] = edges[t]; cnt[t] = 0u; accs[t] = 0u; cfs[t] = 0.0; }
    __syncthreads();

    const int stride = NB_PART * 256;
    for (int i = blockIdx.x * 256 + t; i < N; i += stride) {
        float c = conf[i];
        if (c > e[0]) {                                   // "valid" samples only
            int bin = 0;
#pragma unroll
            for (int j = 1; j <= NBINS; ++j) bin += (e[j] < c); // searchsorted left
            if (bin > NBINS - 1) bin = NBINS - 1;
            atomicAdd(&cnt[bin], 1u);
            atomicAdd(&accs[bin], acc[i]);
            atomicAdd(&cfs[bin], (double)c);
        }
    }
    __syncthreads();
    if (t < 16) {
        float* p = partials + (size_t)blockIdx.x * 48;
        p[t]      = (float)cnt[t];
        p[16 + t] = (float)accs[t];
        p[32 + t] = (float)cfs[t];
    }
}

// ---------------------------------------------------------------- Phase 7 ---
// Single wave. Reduce NB_PART x 48 partials with V_WMMA_F32_16X16X4_F32:
// A = ones(16x4) -> D[m][n] += sum_k B[k][n]; exact in f32. Three column
// groups (counts/acc/conf) use three accumulator chains. Then compute ECE.
__global__ void __launch_bounds__(32)
wmma_finalize_kernel(const float* __restrict__ partials, int nb, float Nf,
                     float* __restrict__ out) {
    const int l    = threadIdx.x;   // 0..31, full wave, EXEC all ones
    const int col  = l & 15;
    const int half = l >> 4;

    v8f c0 = {}, c1 = {}, c2 = {};
    v2f a;  a.x = 1.0f; a.y = 1.0f;

    for (int base = 0; base < nb; base += 4) {
        const float* p0 = partials + (size_t)(base + half) * 48 + col;
        const float* p1 = partials + (size_t)(base + 2 + half) * 48 + col;
        v2f b;
        b.x = p0[0];  b.y = p1[0];      // counts columns
        c0 = __builtin_amdgcn_wmma_f32_16x16x4_f32(false, a, false, b, (short)0, c0, false, false);
        b.x = p0[16]; b.y = p1[16];     // acc columns
        c1 = __builtin_amdgcn_wmma_f32_16x16x4_f32(false, a, false, b, (short)0, c1, false, false);
        b.x = p0[32]; b.y = p1[32];     // conf columns
        c2 = __builtin_amdgcn_wmma_f32_16x16x4_f32(false, a, false, b, (short)0, c2, false, false);
    }

    // D[m][n] identical over m; lanes 0..15 hold (M=0, N=lane) in VGPR 0.
    __shared__ float scnt[16], sacc[16], scf[16];
    if (l < 16) { scnt[l] = c0[0]; sacc[l] = c1[0]; scf[l] = c2[0]; }
    __syncthreads();

    if (l == 0) {
        float ece = 0.0f;
        for (int b2 = 0; b2 < NBINS; ++b2) {
            float cf = scnt[b2];
            if (cf > 0.0f) {
                float denom    = fmaxf(cf, 1.0f);
                float avg_acc  = sacc[b2] / denom;
                float avg_conf = scf[b2] / denom;
                ece += fabsf(avg_conf - avg_acc) * (cf / Nf);
            }
        }
        out[0] = ece;
    }
}

// ---------------------------------------------------------------------------
extern "C" void kernel_launch(void* const* d_in, const int* in_sizes, int n_in,
                              void* d_out, int out_size, void* d_ws, size_t ws_size,
                              hipStream_t stream) {
    const int N = in_sizes[1];
    const int C = in_sizes[0] / N;

    const float* logits = (const float*)d_in[0];
    const int*   labels = (const int*)d_in[1];
    float*       out    = (float*)d_out;

    // workspace layout
    char*     ws       = (char*)d_ws;
    float*    conf     = (float*)ws;                                    // N f32
    unsigned* acc      = (unsigned*)(ws + (size_t)N * 4);               // N u32
    unsigned* hist     = (unsigned*)(ws + (size_t)N * 8);               // 65536 u32
    char*     p        = ws + (size_t)N * 8 + 65536u * 4;
    unsigned* tBucket  = (unsigned*)p;  p += 32 * 4;
    unsigned* tLrank   = (unsigned*)p;  p += 32 * 4;
    float*    tVal     = (float*)p;     p += 32 * 4;
    float*    edges    = (float*)p;     p += 16 * 4;
    float*    partials = (float*)p;                                     // NB_PART*48 f32

    // required order-statistic ranks + f32 interp fracs (mimic jnp.linspace/interp)
    RankParams rp;
    FracParams fp;
    const float delta = (float)N / 15.0f;
    rp.r[0] = 0;  fp.f[0] = 0.0f;  fp.f[15] = 0.0f;
    for (int i = 1; i <= 14; ++i) {
        float pos = (float)i * delta;
        int k = (int)floorf(pos);
        if (k > N - 2) k = N - 2;
        rp.r[2 * i - 1] = k;
        rp.r[2 * i]     = k + 1;
        fp.f[i]         = pos - (float)k;
    }
    rp.r[NTGT - 1] = N - 1;

    hipMemsetAsync(hist, 0, 65536u * 4, stream);

    row_reduce_kernel<<<(N + 7) / 8, 256, 0, stream>>>(logits, labels, conf, acc, N, C);
    hist_hi_kernel<<<(N + 255) / 256, 256, 0, stream>>>(conf, hist, N);
    find_buckets_kernel<<<1, 1024, 0, stream>>>(hist, rp, tBucket, tLrank);
    refine_kernel<<<NTGT, 256, 0, stream>>>(conf, N, tBucket, tLrank, tVal);
    make_edges_kernel<<<1, 32, 0, stream>>>(tVal, edges, fp);
    bin_accum_kernel<<<NB_PART, 256, 0, stream>>>(conf, acc, edges, N, partials);
    wmma_finalize_kernel<<<1, 32, 0, stream>>>(partials, NB_PART, (float)N, out);
}